// SeparateHiddenGCVAEDecoder_16286515987224
// MI455X (gfx1250) — compile-verified
//
#include <hip/hip_runtime.h>
#include <hip/hip_bf16.h>

typedef __attribute__((ext_vector_type(2))) float v2f;
typedef __attribute__((ext_vector_type(8))) float v8f;

#define DOUT 256

__device__ __forceinline__ void atomic_add_f32(float* p, float v) {
  __hip_atomic_fetch_add(p, v, __ATOMIC_RELAXED, __HIP_MEMORY_SCOPE_AGENT);
}

// ---------------------------------------------------------------- zero fill
__global__ void zero_kernel(float* __restrict__ p, long n) {
  long i = (long)blockIdx.x * blockDim.x + threadIdx.x;
  long stride = (long)gridDim.x * blockDim.x;
  for (; i < n; i += stride) p[i] = 0.0f;
}

// ------------------------------------------------- in-degree of dst (float)
__global__ void degree_kernel(const int* __restrict__ dst, int E,
                              float* __restrict__ deg) {
  long i = (long)blockIdx.x * blockDim.x + threadIdx.x;
  long stride = (long)gridDim.x * blockDim.x;
  for (; i < E; i += stride) atomic_add_f32(&deg[dst[i]], 1.0f);
}

// ------------------------------------ dinv = rsqrt(deg + 1) [self loop +1]
__global__ void dinv_kernel(float* __restrict__ deg_dinv, int N) {
  int i = blockIdx.x * blockDim.x + threadIdx.x;
  if (i < N) deg_dinv[i] = rsqrtf(deg_dinv[i] + 1.0f);
}

// ---------------------------------------------------------------- WMMA GEMM
// G = dinv[:,None] * (X @ W),  X:[N,K] row-major, W:[K,256] row-major.
// One wave computes a 16x64 output strip using V_WMMA_F32_16X16X4_F32.
// fp32 A-frag layout (16x4, 2 VGPRs): lane<16 holds M=lane, K={k0,k0+1};
// lane>=16 holds M=lane-16, K={k0+2,k0+3}.  B-frag (4x16) mirrors it.
// fp32 C/D layout: VGPR r, lanes0-15 -> M=r,N=lane; lanes16-31 -> M=r+8.
__global__ __launch_bounds__(256) void gemm_rowscale_kernel(
    const float* __restrict__ X, const float* __restrict__ W,
    const float* __restrict__ dinv, float* __restrict__ G,
    int Nrows, int K) {
  const int lane = threadIdx.x & 31;
  const int wave = threadIdx.x >> 5;
  const int tile = blockIdx.x * 8 + wave;      // 8 waves per block
  const int ntiles = (Nrows >> 4) * 4;         // 4 col-groups of 64
  if (tile >= ntiles) return;
  const int mt = tile >> 2;
  const int ng = tile & 3;
  const int half = lane >> 4;
  const int tl = lane & 15;
  const int row = (mt << 4) + tl;
  const float* __restrict__ xrow = X + (size_t)row * K;
  const int colbase = ng * 64 + tl;

  v8f acc0 = {}, acc1 = {}, acc2 = {}, acc3 = {};
  #pragma unroll 4
  for (int k0 = 0; k0 < K; k0 += 4) {
    const int ka = k0 + 2 * half;
    v2f a;
    a.x = xrow[ka];
    a.y = xrow[ka + 1];
    const float* __restrict__ w0 = W + (size_t)ka * DOUT + colbase;
    const float* __restrict__ w1 = w0 + DOUT;
    v2f b0, b1, b2, b3;
    b0.x = w0[0];  b0.y = w1[0];
    b1.x = w0[16]; b1.y = w1[16];
    b2.x = w0[32]; b2.y = w1[32];
    b3.x = w0[48]; b3.y = w1[48];
    acc0 = __builtin_amdgcn_wmma_f32_16x16x4_f32(false, a, false, b0, (short)0, acc0, false, false);
    acc1 = __builtin_amdgcn_wmma_f32_16x16x4_f32(false, a, false, b1, (short)0, acc1, false, false);
    acc2 = __builtin_amdgcn_wmma_f32_16x16x4_f32(false, a, false, b2, (short)0, acc2, false, false);
    acc3 = __builtin_amdgcn_wmma_f32_16x16x4_f32(false, a, false, b3, (short)0, acc3, false, false);
  }

  #pragma unroll
  for (int r = 0; r < 8; ++r) {
    const int orow = (mt << 4) + r + 8 * half;
    const float s = dinv[orow];
    float* __restrict__ grow = G + (size_t)orow * DOUT + ng * 64 + tl;
    grow[0]  = acc0[r] * s;
    grow[16] = acc1[r] * s;
    grow[32] = acc2[r] * s;
    grow[48] = acc3[r] * s;
  }
}

// -------------------------------------------- edge scatter-add aggregation
// ACC[dst[e], :] += G[src[e], :].  64 consecutive lanes share one edge ->
// coalesced float4 gathers (L2-resident) and coalesced f32 atomics.
__global__ __launch_bounds__(256) void agg_kernel(
    const int* __restrict__ src, const int* __restrict__ dst,
    const float* __restrict__ G, float* __restrict__ ACC, int E) {
  const long total = (long)E * 64;
  long idx = (long)blockIdx.x * blockDim.x + threadIdx.x;
  const long stride = (long)gridDim.x * blockDim.x;
  for (; idx < total; idx += stride) {
    const long e = idx >> 6;
    const int c = (int)(idx & 63) << 2;
    const int s = src[e];
    const int d = dst[e];
    const float4 g = *(const float4*)(G + (size_t)s * DOUT + c);
    float* __restrict__ a = ACC + (size_t)d * DOUT + c;
    atomic_add_f32(a + 0, g.x);
    atomic_add_f32(a + 1, g.y);
    atomic_add_f32(a + 2, g.z);
    atomic_add_f32(a + 3, g.w);
  }
}

// ---------------- out = act(dinv*(ACC + G) + b); +G covers the self loop.
__global__ __launch_bounds__(256) void finalize_kernel(
    const float* __restrict__ ACC, const float* __restrict__ G,
    const float* __restrict__ dinv, const float* __restrict__ bias,
    float* __restrict__ out, int N, int dstStride, int dstOff, int act) {
  const long total = (long)N * DOUT;
  long idx = (long)blockIdx.x * blockDim.x + threadIdx.x;
  const long stride = (long)gridDim.x * blockDim.x;
  for (; idx < total; idx += stride) {
    const long i = idx >> 8;
    const int j = (int)(idx & 255);
    float v = dinv[i] * (ACC[idx] + G[idx]) + bias[j];
    if (act) v = tanhf(v);
    out[(size_t)i * dstStride + dstOff + j] = v;
  }
}

// ---------------------------------------------------------------------------
static void run_layer(const float* X, int K, const float* W, const float* b,
                      const float* dinv, float* G, float* ACC,
                      const int* src, const int* dst, int N, int E,
                      float* out, int dstStride, int dstOff, int act,
                      hipStream_t stream) {
  const long accN = (long)N * DOUT;
  zero_kernel<<<4096, 256, 0, stream>>>(ACC, accN);
  const int waves = (N >> 4) * 4;
  gemm_rowscale_kernel<<<(waves + 7) / 8, 256, 0, stream>>>(X, W, dinv, G, N, K);
  agg_kernel<<<40960, 256, 0, stream>>>(src, dst, G, ACC, E);
  finalize_kernel<<<(int)((accN + 255) / 256), 256, 0, stream>>>(
      ACC, G, dinv, b, out, N, dstStride, dstOff, act);
}

extern "C" void kernel_launch(void* const* d_in, const int* in_sizes, int n_in,
                              void* d_out, int out_size, void* d_ws, size_t ws_size,
                              hipStream_t stream) {
  const float* latent    = (const float*)d_in[0];
  const float* condition = (const float*)d_in[1];
  const int*   edges     = (const int*)d_in[2];
  const float* Wz = (const float*)d_in[3]; const float* bz = (const float*)d_in[4];
  const float* Wc = (const float*)d_in[5]; const float* bc = (const float*)d_in[6];
  const float* Wh = (const float*)d_in[7]; const float* bh = (const float*)d_in[8];
  const float* Wo = (const float*)d_in[9]; const float* bo = (const float*)d_in[10];

  const int N = in_sizes[0] / 256;   // LD = 256
  const int E = in_sizes[2] / 2;
  const int* src = edges;            // edge_index[0,:]
  const int* dst = edges + E;        // edge_index[1,:]

  // Workspace layout (floats): dinv[N] | G[N*256] | ACC[N*256] | Hcat[N*512] | H2[N*256]
  float* ws   = (float*)d_ws;
  float* dinv = ws;
  float* G    = dinv + N;
  float* ACC  = G + (size_t)N * DOUT;
  float* Hcat = ACC + (size_t)N * DOUT;
  float* H2   = Hcat + (size_t)N * 2 * DOUT;
  float* OUT  = (float*)d_out;

  // deg -> dinv (deg stored in dinv buffer, converted in place)
  zero_kernel<<<256, 256, 0, stream>>>(dinv, N);
  degree_kernel<<<1024, 256, 0, stream>>>(dst, E, dinv);
  dinv_kernel<<<(N + 255) / 256, 256, 0, stream>>>(dinv, N);

  // z2h = tanh(gcn(latent, Wz)) -> Hcat[:, 0:256]
  run_layer(latent, 256, Wz, bz, dinv, G, ACC, src, dst, N, E,
            Hcat, 512, 0, 1, stream);
  // c2h = tanh(gcn(condition, Wc)) -> Hcat[:, 256:512]
  run_layer(condition, 128, Wc, bc, dinv, G, ACC, src, dst, N, E,
            Hcat, 512, 256, 1, stream);
  // h = tanh(gcn(Hcat, Wh)) -> H2
  run_layer(Hcat, 512, Wh, bh, dinv, G, ACC, src, dst, N, E,
            H2, 256, 0, 1, stream);
  // out = gcn(H2, Wo) -> d_out
  run_layer(H2, 256, Wo, bo, dinv, G, ACC, src, dst, N, E,
            OUT, 256, 0, 0, stream);
}